// CenterLoss_20426864460160
// MI455X (gfx1250) — compile-verified
//
#include <hip/hip_runtime.h>
#include <stdint.h>

#define NUM_CLASSES 50000
#define FEAT_DIM    256
#define BATCH       131072
#define LAMBDA_CENT 0.5f

#define BLOCK_THREADS   256
#define WAVE_SIZE       32
#define WAVES_PER_BLOCK (BLOCK_THREADS / WAVE_SIZE)
#define MAIN_BLOCKS     2048   // 16384 waves -> 8 samples per wave

typedef int cl_v4i __attribute__((vector_size(16)));

// ---------------- CDNA5 async global->LDS (ASYNCcnt) ----------------
#if defined(__AMDGCN__) && __has_builtin(__builtin_amdgcn_global_load_async_to_lds_b128)
#define ASYNC_LOAD_B128(G, L)                                               \
  __builtin_amdgcn_global_load_async_to_lds_b128(                           \
      (__attribute__((address_space(1))) cl_v4i*)(cl_v4i*)(void*)(G),       \
      (__attribute__((address_space(3))) cl_v4i*)(cl_v4i*)(void*)(L), 0, 0)
#elif defined(__AMDGCN__)
#define ASYNC_LOAD_B128(G, L)                                               \
  do {                                                                      \
    unsigned _ldsoff = (unsigned)(unsigned long long)(const void*)(L);      \
    asm volatile("global_load_async_to_lds_b128 %0, %1, off"                \
                 :: "v"(_ldsoff), "v"((const void*)(G)) : "memory");        \
  } while (0)
#else
#define ASYNC_LOAD_B128(G, L) ((void)(G), (void)(L))
#endif

#if defined(__AMDGCN__) && __has_builtin(__builtin_amdgcn_s_wait_asynccnt)
#define WAIT_ASYNCCNT(N) __builtin_amdgcn_s_wait_asynccnt(N)
#elif defined(__AMDGCN__)
#define WAIT_ASYNCCNT(N) asm volatile("s_wait_asynccnt " #N ::: "memory")
#else
#define WAIT_ASYNCCNT(N) ((void)0)
#endif

// ---------------- kernel 0: per-call init ----------------
// new_centers <- centers (also warms the 51MB table into L2), cnt <- 0, lossAcc <- 0
__global__ __launch_bounds__(BLOCK_THREADS)
void cl_init_kernel(const float* __restrict__ centers,
                    float* __restrict__ out,        // out[0]=loss, out[1..]=new_centers
                    int* __restrict__ cnt,
                    double* __restrict__ lossAcc) {
  const long long N = (long long)NUM_CLASSES * FEAT_DIM;
  long long t = (long long)blockIdx.x * blockDim.x + threadIdx.x;
  const long long stride = (long long)gridDim.x * blockDim.x;
  for (long long k = t; k < N; k += stride) out[1 + k] = centers[k];
  for (long long k = t; k < NUM_CLASSES; k += stride) cnt[k] = 0;
  if (t == 0) *lossAcc = 0.0;
}

// ---------------- kernel 1: label histogram ----------------
__global__ __launch_bounds__(BLOCK_THREADS)
void cl_count_kernel(const int* __restrict__ labels, int* __restrict__ cnt) {
  const int t = blockIdx.x * blockDim.x + threadIdx.x;
  if (t < BATCH) atomicAdd(&cnt[labels[t]], 1);
}

// ---------------- kernel 2: async-pipelined main pass ----------------
// One wave per sample. Per sample: 2x B128 async loads for the feature row,
// 2x for the gathered center row (1KB each), double-buffered across the
// grid-stride loop with s_wait_asynccnt 4 overlapping next-sample fetch.
__global__ __launch_bounds__(BLOCK_THREADS)
void cl_main_kernel(const float* __restrict__ feat,
                    const int*   __restrict__ labels,
                    const float* __restrict__ centers,
                    const int*   __restrict__ cnt,
                    float*       __restrict__ newC,     // = out + 1
                    double*      __restrict__ lossAcc) {
  __shared__ float lds[WAVES_PER_BLOCK][2][2][FEAT_DIM];  // [wave][parity][feat|cent][256] = 32KB
  const int lane   = threadIdx.x & (WAVE_SIZE - 1);
  const int wv     = threadIdx.x / WAVE_SIZE;
  const int waveId = blockIdx.x * WAVES_PER_BLOCK + wv;
  const int nWaves = gridDim.x * WAVES_PER_BLOCK;

  float laneLoss = 0.0f;

  int i = waveId;
  int lab = 0;
  if (i < BATCH) {
    lab = labels[i];
    const float* fr = feat    + (size_t)i   * FEAT_DIM;
    const float* cr = centers + (size_t)lab * FEAT_DIM;
    ASYNC_LOAD_B128(fr +       lane * 4, &lds[wv][0][0][      lane * 4]);
    ASYNC_LOAD_B128(fr + 128 + lane * 4, &lds[wv][0][0][128 + lane * 4]);
    ASYNC_LOAD_B128(cr +       lane * 4, &lds[wv][0][1][      lane * 4]);
    ASYNC_LOAD_B128(cr + 128 + lane * 4, &lds[wv][0][1][128 + lane * 4]);
  }

  int p = 0;
  while (i < BATCH) {
    const int inext = i + nWaves;
    int labn = 0;
    if (inext < BATCH) {
      labn = labels[inext];
      const float* frn = feat    + (size_t)inext * FEAT_DIM;
      const float* crn = centers + (size_t)labn  * FEAT_DIM;
      const int q = p ^ 1;
      ASYNC_LOAD_B128(frn +       lane * 4, &lds[wv][q][0][      lane * 4]);
      ASYNC_LOAD_B128(frn + 128 + lane * 4, &lds[wv][q][0][128 + lane * 4]);
      ASYNC_LOAD_B128(crn +       lane * 4, &lds[wv][q][1][      lane * 4]);
      ASYNC_LOAD_B128(crn + 128 + lane * 4, &lds[wv][q][1][128 + lane * 4]);
      WAIT_ASYNCCNT(4);   // async loads complete in order: current sample's 4 are done
    } else {
      WAIT_ASYNCCNT(0);
    }

    const float4 fa = *(const float4*)&lds[wv][p][0][      lane * 4];
    const float4 fb = *(const float4*)&lds[wv][p][0][128 + lane * 4];
    const float4 ca = *(const float4*)&lds[wv][p][1][      lane * 4];
    const float4 cb = *(const float4*)&lds[wv][p][1][128 + lane * 4];

    // row sum of squares -> wave32 xor-shuffle reduction
    float ss = fa.x*fa.x + fa.y*fa.y + fa.z*fa.z + fa.w*fa.w
             + fb.x*fb.x + fb.y*fb.y + fb.z*fb.z + fb.w*fb.w;
#pragma unroll
    for (int off = WAVE_SIZE / 2; off > 0; off >>= 1)
      ss += __shfl_xor(ss, off, WAVE_SIZE);

    const float inv = 1.0f / fmaxf(sqrtf(ss), 1e-12f);                 // F.normalize eps
    const float wgt = 1.0f / (1.0f + (float)cnt[lab]);                 // 1/(1+count)
    const float upd = LAMBDA_CENT * wgt;                               // new = c + 0.5*w*(f - c)
    float* dst = newC + (size_t)lab * FEAT_DIM;
    const int base = lane * 4;

    float fn, d;
    fn = fa.x*inv; d = fn - ca.x; laneLoss += d*d; atomicAdd(dst + base + 0,       upd * d);
    fn = fa.y*inv; d = fn - ca.y; laneLoss += d*d; atomicAdd(dst + base + 1,       upd * d);
    fn = fa.z*inv; d = fn - ca.z; laneLoss += d*d; atomicAdd(dst + base + 2,       upd * d);
    fn = fa.w*inv; d = fn - ca.w; laneLoss += d*d; atomicAdd(dst + base + 3,       upd * d);
    fn = fb.x*inv; d = fn - cb.x; laneLoss += d*d; atomicAdd(dst + 128 + base + 0, upd * d);
    fn = fb.y*inv; d = fn - cb.y; laneLoss += d*d; atomicAdd(dst + 128 + base + 1, upd * d);
    fn = fb.z*inv; d = fn - cb.z; laneLoss += d*d; atomicAdd(dst + 128 + base + 2, upd * d);
    fn = fb.w*inv; d = fn - cb.w; laneLoss += d*d; atomicAdd(dst + 128 + base + 3, upd * d);

    i = inext;
    lab = labn;
    p ^= 1;
  }

  // per-wave loss total -> one f64 global atomic per wave (16384 total)
#pragma unroll
  for (int off = WAVE_SIZE / 2; off > 0; off >>= 1)
    laneLoss += __shfl_xor(laneLoss, off, WAVE_SIZE);
  if (lane == 0 && waveId < BATCH)
    atomicAdd(lossAcc, (double)laneLoss);
}

// ---------------- kernel 3: finalize loss ----------------
__global__ void cl_finish_kernel(const double* __restrict__ lossAcc,
                                 float* __restrict__ out) {
  if (blockIdx.x == 0 && threadIdx.x == 0)
    out[0] = (float)(*lossAcc * (1.0 / (double)BATCH));
}

extern "C" void kernel_launch(void* const* d_in, const int* in_sizes, int n_in,
                              void* d_out, int out_size, void* d_ws, size_t ws_size,
                              hipStream_t stream) {
  (void)in_sizes; (void)n_in; (void)out_size; (void)ws_size;
  const float* feat    = (const float*)d_in[0];   // [131072, 256] f32
  const int*   labels  = (const int*)  d_in[1];   // [131072] i32
  const float* centers = (const float*)d_in[2];   // [50000, 256] f32
  float* out = (float*)d_out;                     // [0]=loss, [1..]=new_centers

  double* lossAcc = (double*)d_ws;                        // 8B accumulator
  int*    cnt     = (int*)((char*)d_ws + 16);             // 50000 ints

  cl_init_kernel<<<4096, BLOCK_THREADS, 0, stream>>>(centers, out, cnt, lossAcc);
  cl_count_kernel<<<BATCH / BLOCK_THREADS, BLOCK_THREADS, 0, stream>>>(labels, cnt);
  cl_main_kernel<<<MAIN_BLOCKS, BLOCK_THREADS, 0, stream>>>(feat, labels, centers,
                                                            cnt, out + 1, lossAcc);
  cl_finish_kernel<<<1, 64, 0, stream>>>(lossAcc, out);
}